// GNN_59974923321371
// MI455X (gfx1250) — compile-verified
//
#include <hip/hip_runtime.h>
#include <hip/hip_bf16.h>
#include <math.h>

#define N_NODES 30000
#define FDIM    256
#define HDIM    128
#define K2      256      // 2*HDIM == FDIM
#define RREL    3
#define DEG     8
#define NEDGE   (N_NODES*DEG)
#define APAD    260      // 16-row A tile row pitch (floats), avoids LDS bank conflicts
#define ZPAD    132      // Zn/Zc row pitch
#define WMAT    (K2*HDIM) // 32768 floats per weight matrix

typedef float v2f __attribute__((ext_vector_type(2)));
typedef float v8f __attribute__((ext_vector_type(8)));

// ---------------------------------------------------------------------------
// [16,K2] (LDS, pitch APAD) x [K2,128] (transposed weights WT[col][k], global)
// -> 16x16 tile at col0 via V_WMMA_F32_16X16X4_F32, 2-way split-K for WMMA ILP.
// A 16x4 layout: lanes0-15: M=lane, K={0,1}; lanes16-31: M=lane-16, K={2,3}
// B 4x16 layout: v0: lanes0-15 K=0, lanes16-31 K=2; v1: K=1 / K=3 (N=lane&15)
// With WT, each lane's B pair is contiguous -> one 8-byte load per WMMA.
// ---------------------------------------------------------------------------
__device__ __forceinline__ v8f wmma_gemm16T(const float* __restrict__ Alds,
                                            const float* __restrict__ WT,
                                            int col0, v8f acc) {
    const int lane = threadIdx.x & 31;
    const int m    = lane & 15;           // also the N index for B
    const int kb   = (lane >> 4) << 1;    // 0 or 2
    const float* __restrict__ ar = Alds + m * APAD + kb;
    const float* __restrict__ wr = WT + (size_t)(col0 + m) * K2 + kb;
    v8f c0 = acc, c1 = {};
    #pragma unroll 8
    for (int kk = 0; kk < K2 / 2; kk += 4) {
        v2f a0, a1;
        a0.x = ar[kk];            a0.y = ar[kk + 1];
        a1.x = ar[kk + K2 / 2];   a1.y = ar[kk + 1 + K2 / 2];
        v2f b0 = *(const v2f*)(wr + kk);
        v2f b1 = *(const v2f*)(wr + kk + K2 / 2);
        c0 = __builtin_amdgcn_wmma_f32_16x16x4_f32(false, a0, false, b0,
                                                   (short)0, c0, false, false);
        c1 = __builtin_amdgcn_wmma_f32_16x16x4_f32(false, a1, false, b1,
                                                   (short)0, c1, false, false);
    }
    return c0 + c1;
}

// ---------------------------------------------------------------------------
// one-time weight transpose: WT[m][col][k] = W[m][k][col]  (nmat matrices)
// ---------------------------------------------------------------------------
__global__ __launch_bounds__(256)
void transpose_kernel(const float* __restrict__ src, float* __restrict__ dst,
                      int nmat) {
    int total = nmat * WMAT;
    for (int i = blockIdx.x * 256 + threadIdx.x; i < total; i += gridDim.x * 256) {
        int m = i / WMAT, rem = i - m * WMAT;
        int k = rem / HDIM, col = rem - k * HDIM;
        dst[m * WMAT + col * K2 + k] = src[i];
    }
}

// ---------------------------------------------------------------------------
// mask / neighbourhood counts from dst (faithful to reference)
// ---------------------------------------------------------------------------
__global__ __launch_bounds__(256)
void maskcount_kernel(const int* __restrict__ dst, float* counts, float* mask) {
    long i = (long)blockIdx.x * 256 + threadIdx.x;
    if (i < (long)2 * RREL * NEDGE) {
        int d = dst[i];
        mask[d] = 1.0f;
        if (i < (long)RREL * NEDGE) atomicAdd(counts + d, 1.0f);
    }
}

// wvec_r = W1_attnlin[r] @ attnvec[r,H:];  crs_r = b1_attnlin[r].attnvec[r,H:] + b1_attnvec[r]
__global__ __launch_bounds__(256)
void precompute_kernel(const float* __restrict__ W1a, const float* __restrict__ b1a,
                       const float* __restrict__ avec, const float* __restrict__ bvec,
                       float* wvec, float* crs) {
    int tid = blockIdx.x * 256 + threadIdx.x;
    for (int p = tid; p < RREL * K2; p += gridDim.x * 256) {
        int r = p / K2, f = p - r * K2;
        const float* Wrow = W1a + ((size_t)r * K2 + f) * HDIM;
        const float* av   = avec + r * K2 + HDIM;
        float s = 0.f;
        for (int j = 0; j < HDIM; ++j) s += Wrow[j] * av[j];
        wvec[p] = s;
    }
    if (tid < RREL) {
        const float* av = avec + tid * K2 + HDIM;
        float s = 0.f;
        for (int j = 0; j < HDIM; ++j) s += b1a[tid * HDIM + j] * av[j];
        crs[tid] = s + bvec[tid];
    }
}

// ---------------------------------------------------------------------------
// layer 0: h[n,:128] = feat@W0_self+b0;  h[n,128:] = (Σ_r agg_r@W0_rel[r] + 8Σb)/nsz
// 16 nodes per block; aggregation pre-GEMM (linear commutes with segment sum,
// dst is repeat(arange(N),DEG) so node n owns edges [n*8, n*8+8)).
// ---------------------------------------------------------------------------
__global__ __launch_bounds__(256)
void layer0_kernel(const float* __restrict__ feat, const int* __restrict__ src,
                   const float* __restrict__ W0sT, const float* __restrict__ b0s,
                   const float* __restrict__ W0rT, const float* __restrict__ b0r,
                   const float* __restrict__ counts, const float* __restrict__ mask,
                   float* __restrict__ h) {
    __shared__ float Aself[16 * APAD];
    __shared__ float Agg[16 * APAD];
    const int tid  = threadIdx.x;
    const int n0   = blockIdx.x * 16;
    const int row  = tid >> 4, t16 = tid & 15;
    const int wave = tid >> 5, lane = tid & 31;
    const int col0 = wave * 16;

    // stage self tile [16,256]
    {
        const float4* sr = (const float4*)(feat + (size_t)(n0 + row) * FDIM);
        float4* dr = (float4*)(Aself + row * APAD);
        #pragma unroll
        for (int i = 0; i < 4; ++i) dr[t16 * 4 + i] = sr[t16 * 4 + i];
    }
    __syncthreads();
    v8f cs = {};
    cs = wmma_gemm16T(Aself, W0sT, col0, cs);

    v8f cr = {};
    for (int r = 0; r < RREL; ++r) {
        __syncthreads();   // previous Agg consumers done
        // gather-sum 8 neighbour feat rows per tile row
        float4 acc[4] = {};
        const int* sp = src + (size_t)r * NEDGE + (size_t)(n0 + row) * DEG;
        #pragma unroll
        for (int j = 0; j < DEG; ++j) {
            const float4* sr = (const float4*)(feat + (size_t)sp[j] * FDIM);
            #pragma unroll
            for (int i = 0; i < 4; ++i) {
                float4 v = sr[t16 * 4 + i];
                acc[i].x += v.x; acc[i].y += v.y; acc[i].z += v.z; acc[i].w += v.w;
            }
        }
        float4* dr = (float4*)(Agg + row * APAD);
        #pragma unroll
        for (int i = 0; i < 4; ++i) dr[t16 * 4 + i] = acc[i];
        __syncthreads();
        cr = wmma_gemm16T(Agg, W0rT + (size_t)r * WMAT, col0, cr);
    }

    const int colc = col0 + (lane & 15);
    const float bself = b0s[colc];
    const float brel  = (float)DEG * (b0r[colc] + b0r[HDIM + colc] + b0r[2 * HDIM + colc]);
    #pragma unroll
    for (int j = 0; j < 8; ++j) {
        int rn = n0 + j + ((lane >> 4) << 3);
        float mk = mask[rn];
        float ns = fmaxf(counts[rn], 1.0f);
        h[(size_t)rn * K2 + colc]        = (cs[j] + bself) * mk;
        h[(size_t)rn * K2 + HDIM + colc] = ((cr[j] + brel) / ns) * mk;
    }
}

// ---------------------------------------------------------------------------
// self_h = h @ W1_self + b1_self ; also emits out[:, :128] = sigmoid(self_h)*mask
// ---------------------------------------------------------------------------
__global__ __launch_bounds__(256)
void selfh_kernel(const float* __restrict__ h,
                  const float* __restrict__ W1sT, const float* __restrict__ b1s,
                  const float* __restrict__ mask,
                  float* __restrict__ self_h, float* __restrict__ out) {
    __shared__ float A[16 * APAD];
    const int tid = threadIdx.x;
    const int n0  = blockIdx.x * 16;
    const int row = tid >> 4, t16 = tid & 15;
    const int wave = tid >> 5, lane = tid & 31;
    const int col0 = wave * 16;
    {
        const float4* sr = (const float4*)(h + (size_t)(n0 + row) * K2);
        float4* dr = (float4*)(A + row * APAD);
        #pragma unroll
        for (int i = 0; i < 4; ++i) dr[t16 * 4 + i] = sr[t16 * 4 + i];
    }
    __syncthreads();
    v8f c = {};
    c = wmma_gemm16T(A, W1sT, col0, c);
    const int colc = col0 + (lane & 15);
    const float b = b1s[colc];
    #pragma unroll
    for (int j = 0; j < 8; ++j) {
        int rn = n0 + j + ((lane >> 4) << 3);
        float v = c[j] + b;
        self_h[(size_t)rn * HDIM + colc] = v;
        out[(size_t)rn * K2 + colc] = (1.0f / (1.0f + expf(-v))) * mask[rn];
    }
}

// ---------------------------------------------------------------------------
// layer 1 fused: 2 destination nodes per block -> per relation a 16-row tile
// (2 nodes x 8 edges). Zn/Zc via WMMA, scores, pairwise L2, softmax, agg.
// ---------------------------------------------------------------------------
__global__ __launch_bounds__(256)
void layer1_kernel(const float* __restrict__ h, const int* __restrict__ src,
                   const float* __restrict__ W1rT, const float* __restrict__ b1r,
                   const float* __restrict__ W1aT, const float* __restrict__ b1a,
                   const float* __restrict__ avec,
                   const float* __restrict__ wvec, const float* __restrict__ crs,
                   const float* __restrict__ self_h, const float* __restrict__ mask,
                   float* __restrict__ out) {
    __shared__ float Hn[16 * APAD];
    __shared__ float Hc[16 * APAD];
    __shared__ float Zn[48 * ZPAD];   // [node*24 + k][128]
    __shared__ float Zc[48 * ZPAD];
    __shared__ float sh[2 * HDIM];
    __shared__ float selfdot[6];      // [node*3 + r]
    __shared__ float Spart[16];
    __shared__ float Sl[48];          // attention scores S
    __shared__ float rn2[48], rc2[48], dsum[48], gamma[48];

    const int tid = threadIdx.x;
    const int n0  = blockIdx.x * 2;
    const int row = tid >> 4, t16 = tid & 15;
    const int wave = tid >> 5, lane = tid & 31;
    const int col0 = wave * 16;

    if (tid < 2 * HDIM) sh[tid] = self_h[(size_t)n0 * HDIM + tid];
    __syncthreads();
    if (tid < 6) {   // (self_h[dst] . attnvec[r,:H]) per (node, relation)
        int node = tid / 3, r = tid - node * 3;
        const float* av = avec + r * K2;
        float s = 0.f;
        for (int j = 0; j < HDIM; ++j) s += sh[node * HDIM + j] * av[j];
        selfdot[tid] = s;
    }

    for (int r = 0; r < RREL; ++r) {
        __syncthreads();   // previous-iteration consumers of Hn/Hc/Spart done
        {   // gather h rows: tile row t -> node n0+(t>>3), edge (t&7)
            int nd = n0 + (row >> 3), j = row & 7;
            int sn = src[(size_t)r * NEDGE + (size_t)nd * DEG + j];
            int sc = src[(size_t)(RREL + r) * NEDGE + (size_t)nd * DEG + j];
            const float4* an = (const float4*)(h + (size_t)sn * K2);
            const float4* ac = (const float4*)(h + (size_t)sc * K2);
            float4* dn = (float4*)(Hn + row * APAD);
            float4* dc = (float4*)(Hc + row * APAD);
            #pragma unroll
            for (int i = 0; i < 4; ++i) { dn[t16 * 4 + i] = an[t16 * 4 + i];
                                          dc[t16 * 4 + i] = ac[t16 * 4 + i]; }
        }
        if (tid < 16) Spart[tid] = 0.f;
        __syncthreads();
        {   // per-edge score partial: h_src . wvec_r
            const float* wv = wvec + r * K2;
            float p = 0.f;
            #pragma unroll
            for (int i = 0; i < 16; ++i)
                p += Hn[row * APAD + t16 * 16 + i] * wv[t16 * 16 + i];
            atomicAdd(&Spart[row], p);
        }
        v8f zn = {}, zc = {};
        zn = wmma_gemm16T(Hn, W1rT + (size_t)r * WMAT, col0, zn);
        zc = wmma_gemm16T(Hc, W1aT + (size_t)r * WMAT, col0, zc);
        const int colc = col0 + (lane & 15);
        const float bn = b1r[r * HDIM + colc];
        const float bc = b1a[r * HDIM + colc];
        #pragma unroll
        for (int j = 0; j < 8; ++j) {
            int t = j + ((lane >> 4) << 3);
            int rz = (t >> 3) * 24 + r * 8 + (t & 7);
            Zn[rz * ZPAD + colc] = zn[j] + bn;
            Zc[rz * ZPAD + colc] = zc[j] + bc;
        }
        __syncthreads();
        if (tid < 16) {
            int node = tid >> 3;
            Sl[node * 24 + r * 8 + (tid & 7)] =
                Spart[tid] + selfdot[node * 3 + r] + crs[r];
        }
    }
    __syncthreads();

    if (tid < 48) {                 // |Zn_k|^2, zero dsum
        const float* zr = Zn + tid * ZPAD;
        float s = 0.f;
        for (int i = 0; i < HDIM; ++i) s += zr[i] * zr[i];
        rn2[tid] = s; dsum[tid] = 0.f;
    } else if (tid < 96) {          // |Zc_c|^2
        int t = tid - 48;
        const float* zr = Zc + t * ZPAD;
        float s = 0.f;
        for (int i = 0; i < HDIM; ++i) s += zr[i] * zr[i];
        rc2[t] = s;
    }
    __syncthreads();

    for (int p = tid; p < 2 * 24 * 24; p += 256) {
        int node = p / 576, rem = p - node * 576;
        int k = rem / 24, c = rem - k * 24;
        const float* a = Zn + (node * 24 + k) * ZPAD;
        const float* b = Zc + (node * 24 + c) * ZPAD;
        float dp = 0.f;
        for (int i = 0; i < HDIM; ++i) dp += a[i] * b[i];
        float sq = rn2[node * 24 + k] + rc2[node * 24 + c] - 2.0f * dp;
        atomicAdd(&dsum[node * 24 + k], sqrtf(fmaxf(sq, 1e-12f)));
    }
    __syncthreads();

    if (tid < 2) {                  // softmax alpha (over -dsum) * softmax beta (over S)
        int base = tid * 24;
        float m1 = -1e30f, m2 = -1e30f;
        for (int k = 0; k < 24; ++k) {
            m1 = fmaxf(m1, -dsum[base + k]);
            m2 = fmaxf(m2, Sl[base + k]);
        }
        float s1 = 0.f, s2 = 0.f;
        for (int k = 0; k < 24; ++k) {
            s1 += expf(-dsum[base + k] - m1);
            s2 += expf(Sl[base + k] - m2);
        }
        for (int k = 0; k < 24; ++k)
            gamma[base + k] = (expf(-dsum[base + k] - m1) / s1) *
                              (expf(Sl[base + k] - m2) / s2);
    }
    __syncthreads();

    {   // z_agg1 and output second half
        int node = tid >> 7, colc = tid & 127;
        float z = 0.f;
        #pragma unroll
        for (int k = 0; k < 24; ++k)
            z += gamma[node * 24 + k] * Zn[(node * 24 + k) * ZPAD + colc];
        float mk = mask[n0 + node];
        out[(size_t)(n0 + node) * K2 + HDIM + colc] =
            (1.0f / (1.0f + expf(-z))) * mk;
    }
}

// ---------------------------------------------------------------------------
extern "C" void kernel_launch(void* const* d_in, const int* in_sizes, int n_in,
                              void* d_out, int out_size, void* d_ws, size_t ws_size,
                              hipStream_t stream) {
    const float* feat = (const float*)d_in[0];
    const int*   src  = (const int*)  d_in[1];
    const int*   dst  = (const int*)  d_in[2];
    const float* W0s  = (const float*)d_in[3];
    const float* b0s  = (const float*)d_in[4];
    const float* W0r  = (const float*)d_in[5];
    const float* b0r  = (const float*)d_in[6];
    const float* W1s  = (const float*)d_in[7];
    const float* b1s  = (const float*)d_in[8];
    const float* W1r  = (const float*)d_in[9];
    const float* b1r  = (const float*)d_in[10];
    const float* W1a  = (const float*)d_in[11];
    const float* b1a  = (const float*)d_in[12];
    const float* avec = (const float*)d_in[13];
    const float* bvec = (const float*)d_in[14];
    float* out = (float*)d_out;

    float* ws     = (float*)d_ws;
    float* h      = ws;                                   // N*256
    float* self_h = h + (size_t)N_NODES * K2;             // N*128
    float* counts = self_h + (size_t)N_NODES * HDIM;      // N
    float* mask   = counts + N_NODES;                     // N
    float* wvec   = mask + N_NODES;                       // 3*256
    float* crs    = wvec + RREL * K2;                     // 3
    float* W0sT   = crs + 16;                             // 1 matrix
    float* W0rT   = W0sT + WMAT;                          // 3 matrices
    float* W1sT   = W0rT + 3 * WMAT;                      // 1 matrix
    float* W1rT   = W1sT + WMAT;                          // 3 matrices
    float* W1aT   = W1rT + 3 * WMAT;                      // 3 matrices

    hipMemsetAsync(counts, 0, 2 * (size_t)N_NODES * sizeof(float), stream);

    transpose_kernel<<<128, 256, 0, stream>>>(W0s, W0sT, 1);
    transpose_kernel<<<384, 256, 0, stream>>>(W0r, W0rT, 3);
    transpose_kernel<<<128, 256, 0, stream>>>(W1s, W1sT, 1);
    transpose_kernel<<<384, 256, 0, stream>>>(W1r, W1rT, 3);
    transpose_kernel<<<384, 256, 0, stream>>>(W1a, W1aT, 3);

    maskcount_kernel<<<(2 * RREL * NEDGE + 255) / 256, 256, 0, stream>>>(dst, counts, mask);
    precompute_kernel<<<3, 256, 0, stream>>>(W1a, b1a, avec, bvec, wvec, crs);
    layer0_kernel<<<N_NODES / 16, 256, 0, stream>>>(feat, src, W0sT, b0s, W0rT, b0r,
                                                    counts, mask, h);
    selfh_kernel<<<N_NODES / 16, 256, 0, stream>>>(h, W1sT, b1s, mask, self_h, out);
    layer1_kernel<<<N_NODES / 2, 256, 0, stream>>>(h, src, W1rT, b1r, W1aT, b1a, avec,
                                                   wvec, crs, self_h, mask, out);
}